// inf_loss_19937238188081
// MI455X (gfx1250) — compile-verified
//
#include <hip/hip_runtime.h>

#define W 384
#define H 384
#define PLANE 147456            // 384*384
#define NPLANES 64              // B*C = 2*32
#define NYB 8                   // y-bands per plane in loss kernel
#define NPART (NPLANES * NYB)   // 512 deterministic partial sums
#define EPSF 1e-6f

typedef __attribute__((ext_vector_type(2))) float v2f;
typedef __attribute__((ext_vector_type(8))) float v8f;

#if __has_builtin(__builtin_amdgcn_wmma_f32_16x16x4_f32)
#define HAVE_WMMA_F32 1
#else
#define HAVE_WMMA_F32 0
#warning "CDNA5 builtin __builtin_amdgcn_wmma_f32_16x16x4_f32 not available; K2 uses VALU fallback scan"
#endif

// ---------------------------------------------------------------- K0: zero partials
__global__ void k0_zero(float* __restrict__ p) {
    for (int i = threadIdx.x; i < NPART; i += blockDim.x) p[i] = 0.f;
}

// ---------------------------------------------------------------- K1: build 8 fields
// (shifted by -0.5: var/cov are shift invariant, keeps SAT magnitudes ~sqrt(N))
// and do the COLUMN inclusive prefix sum. Lane-owns-column => every access coalesced,
// running sum in a register, no shuffles needed.
// Field order: 0:img 1:v 2:i 3:img^2 4:v^2 5:i^2 6:img*v 7:img*i
__global__ void k1_fields_colscan(const float* __restrict__ vv,
                                  const float* __restrict__ ii,
                                  const float* __restrict__ gg,
                                  float* __restrict__ sat,
                                  int np, int p0) {
    int pl = blockIdx.x >> 2;          // chunk-local plane
    int fq = blockIdx.x & 3;           // field pair 2*fq, 2*fq+1
    int g  = p0 + pl;                  // global plane
    int x  = threadIdx.x;              // 0..383 (block = 384 = 12 waves)

    const float* V = vv + (size_t)g * PLANE;
    const float* I = ii + (size_t)g * PLANE;
    const float* G = gg + (size_t)g * PLANE;
    float* s0 = sat + ((size_t)(2 * fq + 0) * np + pl) * PLANE;
    float* s1 = sat + ((size_t)(2 * fq + 1) * np + pl) * PLANE;

    float c0 = 0.f, c1 = 0.f;
    for (int y = 0; y < H; ++y) {
        int idx = y * W + x;
        float gx = G[idx] - 0.5f;
        float vx = V[idx] - 0.5f;
        float ix = I[idx] - 0.5f;
        float f0, f1;
        switch (fq) {
            case 0:  f0 = gx;      f1 = vx;      break;
            case 1:  f0 = ix;      f1 = gx * gx; break;
            case 2:  f0 = vx * vx; f1 = ix * ix; break;
            default: f0 = gx * vx; f1 = gx * ix; break;
        }
        c0 += f0; c1 += f1;
        s0[idx] = c0;
        s1[idx] = c1;
    }
}

// ---------------------------------------------------------------- K2: ROW inclusive
// prefix sum via WMMA: for a 16x16 tile X, X*U (U = upper-triangular ones) is the
// row-inclusive prefix; f32 WMMA is 16x16x4, so 4 chained WMMAs per tile. The
// inter-tile carry (= D[M][15]) is moved with a single cross-lane shuffle from
// lane 15 (rows 0-7) / lane 31 (rows 8-15) straight into the next accumulator
// init — zero LDS traffic, zero barriers, EXEC stays all-ones around the WMMAs.
// One wave per (field, plane, 16-row band); 8 independent waves per block.
__global__ void k2_rowscan_wmma(float* __restrict__ sat, int np) {
    int wave = threadIdx.x >> 5;
    int lane = threadIdx.x & 31;
    int unit = blockIdx.x * 8 + wave;          // 192*np units total, grid = 24*np
    int f    = unit / (np * 24);
    int rem  = unit % (np * 24);
    int pl   = rem / 24;
    int band = rem % 24;

    float* base = sat + ((size_t)f * np + pl) * PLANE + (size_t)(band * 16) * W;

#if HAVE_WMMA_F32
    // A layout (16x4 f32): lanes 0-15 hold M=lane, K={0,1}; lanes 16-31 M=lane-16, K={2,3}
    int mrow = (lane < 16) ? lane : (lane - 16);
    int koff = (lane < 16) ? 0 : 2;
    // C/D layout: VGPR r -> lanes 0-15: M=r, N=lane; lanes 16-31: M=r+8, N=lane-16
    int ncol = lane & 15;
    int mc   = (lane < 16) ? 0 : 8;
    // B layout (4x16): VGPR0 rows {K0,K1} across lane halves, VGPR1 rows {K2,K3}
    int kr0 = (lane < 16) ? 0 : 1;
    int kr1 = (lane < 16) ? 2 : 3;
    v2f bfr[4];
#pragma unroll
    for (int t = 0; t < 4; ++t) {
        bfr[t].x = ((4 * t + kr0) <= ncol) ? 1.f : 0.f;   // U[k][n] = (k <= n)
        bfr[t].y = ((4 * t + kr1) <= ncol) ? 1.f : 0.f;
    }

    int carrySrc = (lane < 16) ? 15 : 31;      // lane holding D[M][15] for this half
    v8f c;
#pragma unroll
    for (int r = 0; r < 8; ++r) c[r] = 0.f;    // carry starts at zero

    for (int xt = 0; xt < 24; ++xt) {
        const float* rp = base + (size_t)mrow * W + xt * 16;
        v2f a0 = { rp[koff + 0],  rp[koff + 1]  };
        v2f a1 = { rp[koff + 4],  rp[koff + 5]  };
        v2f a2 = { rp[koff + 8],  rp[koff + 9]  };
        v2f a3 = { rp[koff + 12], rp[koff + 13] };

        v8f d = c;                              // accumulator pre-loaded with carry[M]
        d = __builtin_amdgcn_wmma_f32_16x16x4_f32(false, a0, false, bfr[0], (short)0, d, false, false);
        d = __builtin_amdgcn_wmma_f32_16x16x4_f32(false, a1, false, bfr[1], (short)0, d, false, false);
        d = __builtin_amdgcn_wmma_f32_16x16x4_f32(false, a2, false, bfr[2], (short)0, d, false, false);
        d = __builtin_amdgcn_wmma_f32_16x16x4_f32(false, a3, false, bfr[3], (short)0, d, false, false);

#pragma unroll
        for (int r = 0; r < 8; ++r)
            base[(size_t)(mc + r) * W + xt * 16 + ncol] = d[r];   // coalesced per VGPR

#pragma unroll
        for (int r = 0; r < 8; ++r)             // carry for next tile via shuffle
            c[r] = __shfl(d[r], carrySrc, 32);
    }
#else
    if (lane < 16) {
        float run = 0.f;
        float* rp = base + (size_t)lane * W;
        for (int x = 0; x < W; ++x) { run += rp[x]; rp[x] = run; }
    }
#endif
}

// ---------------------------------------------------------------- K3: loss from SAT.
// Lane owns a column; clamped SAT-corner reads are coalesced across lanes. Block
// handles (plane, 48-row band); deterministic hierarchical reduction (no atomics).
__global__ void k3_loss(const float* __restrict__ sat, float* __restrict__ partials,
                        int np, int p0) {
    int pl = blockIdx.x / NYB;
    int yb = blockIdx.x % NYB;
    int g  = p0 + pl;
    int x  = threadIdx.x;

    const int ks[5] = { 3, 7, 15, 31, 61 };
    size_t fstride = (size_t)np * PLANE;
    const float* S0 = sat + (size_t)pl * PLANE;

    float acc = 0.f;
    int y0 = yb * (H / NYB), y1e = y0 + (H / NYB);
    for (int y = y0; y < y1e; ++y) {
        {
            int yp = (y + 4 < H) ? (y + 4) : (H - 1);
            __builtin_prefetch(S0 + (size_t)yp * W + x);          // global_prefetch_b8
        }
#pragma unroll
        for (int s = 0; s < 5; ++s) {
            int k = ks[s], p = k >> 1;
            float inv = 1.f / ((float)k * (float)k);
            int y2 = y + p; if (y2 > H - 1) y2 = H - 1;
            int x2 = x + p; if (x2 > W - 1) x2 = W - 1;
            int ylo = y - p - 1, xlo = x - p - 1;
            bool hy = (ylo >= 0), hx = (xlo >= 0);

            float bs[8];
#pragma unroll
            for (int f = 0; f < 8; ++f) {
                const float* S = S0 + (size_t)f * fstride;
                float a = S[(size_t)y2 * W + x2];
                float b = hy ? S[(size_t)ylo * W + x2] : 0.f;
                float c = hx ? S[(size_t)y2 * W + xlo] : 0.f;
                float d = (hy && hx) ? S[(size_t)ylo * W + xlo] : 0.f;
                bs[f] = (a - b - c + d) * inv;
            }
            float mux = bs[0], muv = bs[1], mui = bs[2];
            float varx = fmaxf(bs[3] - mux * mux, 0.f);
            float varv = fmaxf(bs[4] - muv * muv, 0.f);
            float vari = fmaxf(bs[5] - mui * mui, 0.f);
            float covv = bs[6] - mux * muv;
            float covi = bs[7] - mux * mui;
            acc += covv * rsqrtf(varx * varv + EPSF) + covi * rsqrtf(varx * vari + EPSF);
        }
    }

    // wave32 reduction, then cross-wave via LDS (block = 384 = 12 waves)
    for (int off = 16; off; off >>= 1) acc += __shfl_down(acc, off, 32);
    __shared__ float wsum[12];
    if ((threadIdx.x & 31) == 0) wsum[threadIdx.x >> 5] = acc;
    __syncthreads();
    if (threadIdx.x == 0) {
        float t = 0.f;
        for (int wv = 0; wv < 12; ++wv) t += wsum[wv];
        partials[g * NYB + yb] += t;   // zeroed by K0; one writer per slot => deterministic
    }
}

// ---------------------------------------------------------------- K4: final scalar
__global__ void k4_final(const float* __restrict__ partials, float* __restrict__ out) {
    float s = 0.f;
    for (int i = threadIdx.x; i < NPART; i += 256) s += partials[i];
    for (int off = 16; off; off >>= 1) s += __shfl_down(s, off, 32);
    __shared__ float wsum[8];
    if ((threadIdx.x & 31) == 0) wsum[threadIdx.x >> 5] = s;
    __syncthreads();
    if (threadIdx.x == 0) {
        float t = 0.f;
        for (int w = 0; w < 8; ++w) t += wsum[w];
        // loss = sum_k mean(1 - 0.5*(cc_v+cc_i)) = 5 - 0.5 * sum(cc)/Npix
        out[0] = 5.f - 0.5f * t / 9437184.f;
    }
}

// ---------------------------------------------------------------- host entry
extern "C" void kernel_launch(void* const* d_in, const int* in_sizes, int n_in,
                              void* d_out, int out_size, void* d_ws, size_t ws_size,
                              hipStream_t stream) {
    const float* v    = (const float*)d_in[0];
    const float* ivar = (const float*)d_in[1];
    const float* img  = (const float*)d_in[2];
    float* out = (float*)d_out;

    float* partials = (float*)d_ws;
    const size_t satOff = 4096;
    float* sat = (float*)((char*)d_ws + satOff);

    const size_t perPlaneSAT = (size_t)8 * PLANE * sizeof(float);   // 4.72 MB
    int maxpl = 1;
    if (ws_size > satOff + perPlaneSAT)
        maxpl = (int)((ws_size - satOff) / perPlaneSAT);
    if (maxpl > 32) maxpl = 32;        // cap: 32-plane SAT set = 151 MB < 192 MB L2
    if (maxpl < 1)  maxpl = 1;

    hipLaunchKernelGGL(k0_zero, dim3(1), dim3(256), 0, stream, partials);
    for (int p0 = 0; p0 < NPLANES; p0 += maxpl) {
        int np = NPLANES - p0; if (np > maxpl) np = maxpl;
        hipLaunchKernelGGL(k1_fields_colscan, dim3(np * 4),  dim3(384), 0, stream,
                           v, ivar, img, sat, np, p0);
        hipLaunchKernelGGL(k2_rowscan_wmma,   dim3(np * 24), dim3(256), 0, stream, sat, np);
        hipLaunchKernelGGL(k3_loss,           dim3(np * NYB), dim3(384), 0, stream,
                           sat, partials, np, p0);
    }
    hipLaunchKernelGGL(k4_final, dim3(1), dim3(256), 0, stream, partials, out);
}